// GNNODEFunc_87909390615185
// MI455X (gfx1250) — compile-verified
//
#include <hip/hip_runtime.h>
#include <hip/hip_bf16.h>

typedef __attribute__((ext_vector_type(2))) float v2f;
typedef __attribute__((ext_vector_type(8))) float v8f;

#define NF 64  // hidden width

// ---------------------------------------------------------------- utilities
__global__ void zero_kernel(float* __restrict__ p, long long n) {
    long long i = (long long)blockIdx.x * blockDim.x + threadIdx.x;
    if (i < n) p[i] = 0.0f;
}

__global__ void build_h0_kernel(const float* __restrict__ x,
                                const float* __restrict__ dtr,
                                const float* __restrict__ npd,
                                const float* __restrict__ kp,
                                const float* __restrict__ kpp,
                                const float* __restrict__ imin,
                                const float* __restrict__ imax,
                                float* __restrict__ h0, int N) {
    int i = blockIdx.x * blockDim.x + threadIdx.x;
    if (i >= N) return;
    float* o = h0 + (size_t)i * 8;
    o[0] = x[2 * i + 0];
    o[1] = x[2 * i + 1];
    o[2] = dtr[i];
    o[3] = npd[i];
    o[4] = kp[i];
    o[5] = kpp[i];
    o[6] = imin[i];
    o[7] = imax[i];
}

// weighted + unweighted in-degree in one pass over edges
__global__ void deg_kernel(const int* __restrict__ cols,
                           const float* __restrict__ ed,
                           float* __restrict__ degW,
                           float* __restrict__ degO, int E) {
    int e = blockIdx.x * blockDim.x + threadIdx.x;
    if (e >= E) return;
    int c = cols[e];
    float w = 1.0f / (ed[e] + 1e-6f);
    unsafeAtomicAdd(&degW[c], w);
    unsafeAtomicAdd(&degO[c], 1.0f);
}

// deg -> rsqrt(deg + selfloop) in place (deg+1 >= 1 so no zero guard needed)
__global__ void dinv_kernel(float* __restrict__ degW, float* __restrict__ degO, int N) {
    int i = blockIdx.x * blockDim.x + threadIdx.x;
    if (i >= N) return;
    degW[i] = rsqrtf(degW[i] + 1.0f);
    degO[i] = rsqrtf(degO[i] + 1.0f);
}

// ------------------------------------------------- WMMA GEMM  XW = H x W
// H: [N, K] row-major, W: [K, 64] row-major, XW: [N, 64].
// One wave per 16-row tile; 4 accumulators cover all 64 output columns.
// Fragment layouts per CDNA5 ISA 7.12.2 (32-bit A 16x4 / B 4x16 / C 16x16).
template <int K>
__global__ void gemm_wmma_kernel(const float* __restrict__ H,
                                 const float* __restrict__ W,
                                 float* __restrict__ XW, int ntiles) {
    int wave = blockIdx.x * (blockDim.x >> 5) + (threadIdx.x >> 5);
    if (wave >= ntiles) return;  // wave-uniform: EXEC stays all-ones for WMMA
    int lane = threadIdx.x & 31;
    int m = lane & 15;           // row within tile (A) / col within ntile (B)
    int kb = (lane >> 4) << 1;   // lanes 0-15 -> K+0,K+1 ; lanes 16-31 -> K+2,K+3
    int rowbase = wave * 16;

    v8f acc0 = {}, acc1 = {}, acc2 = {}, acc3 = {};
#pragma unroll
    for (int kk = 0; kk < K; kk += 4) {
        const float* hp = H + (size_t)(rowbase + m) * K + kk + kb;
        v2f a;
        a.x = hp[0];
        a.y = hp[1];
        const float* wp = W + (size_t)(kk + kb) * NF;  // row kk+kb of W
        v2f b0, b1, b2, b3;
        b0.x = wp[m + 0];  b0.y = wp[NF + m + 0];
        b1.x = wp[m + 16]; b1.y = wp[NF + m + 16];
        b2.x = wp[m + 32]; b2.y = wp[NF + m + 32];
        b3.x = wp[m + 48]; b3.y = wp[NF + m + 48];
        acc0 = __builtin_amdgcn_wmma_f32_16x16x4_f32(false, a, false, b0, (short)0, acc0, false, false);
        acc1 = __builtin_amdgcn_wmma_f32_16x16x4_f32(false, a, false, b1, (short)0, acc1, false, false);
        acc2 = __builtin_amdgcn_wmma_f32_16x16x4_f32(false, a, false, b2, (short)0, acc2, false, false);
        acc3 = __builtin_amdgcn_wmma_f32_16x16x4_f32(false, a, false, b3, (short)0, acc3, false, false);
    }
    // C layout: VGPR v -> row v (lanes 0-15) or v+8 (lanes 16-31), col = lane&15
    int rstore = rowbase + ((lane >> 4) << 3);
#pragma unroll
    for (int v = 0; v < 8; ++v) {
        float* o = XW + (size_t)(rstore + v) * NF + m;
        o[0]  = acc0[v];
        o[16] = acc1[v];
        o[32] = acc2[v];
        o[48] = acc3[v];
    }
}

// ------------------------------------------ edge scatter: agg[col] += norm*xw[row]
// 16 lanes per edge, each handles 4 consecutive features (float4 gather).
__global__ void scatter_kernel(const int* __restrict__ rows,
                               const int* __restrict__ cols,
                               const float* __restrict__ ed,
                               const float* __restrict__ dinv,
                               const float* __restrict__ XW,
                               float* __restrict__ AGG,
                               int E, int useWeight) {
    long long j = (long long)blockIdx.x * blockDim.x + threadIdx.x;
    if (j >= (long long)E * 16) return;
    int e = (int)(j >> 4);
    int l = (int)(j & 15);
    int r = rows[e];
    int c = cols[e];
    float w = useWeight ? (1.0f / (ed[e] + 1e-6f)) : 1.0f;
    float nrm = dinv[r] * w * dinv[c];
    const float4 xv = *(const float4*)(XW + (size_t)r * NF + l * 4);
    float* ap = AGG + (size_t)c * NF + l * 4;
    unsafeAtomicAdd(ap + 0, nrm * xv.x);
    unsafeAtomicAdd(ap + 1, nrm * xv.y);
    unsafeAtomicAdd(ap + 2, nrm * xv.z);
    unsafeAtomicAdd(ap + 3, nrm * xv.w);
}

__device__ __forceinline__ float softplus_f(float x) {
    return fmaxf(x, 0.0f) + log1pf(expf(-fabsf(x)));
}

// epilogue: h = softplus(agg + dinv^2 * xw (self loop) + bias)
__global__ void act_kernel(const float* __restrict__ AGG,
                           const float* __restrict__ XW,
                           const float* __restrict__ dinv,
                           const float* __restrict__ bias,
                           float* __restrict__ H, int N) {
    long long idx = (long long)blockIdx.x * blockDim.x + threadIdx.x;
    if (idx >= (long long)N * NF) return;
    int i = (int)(idx >> 6);
    int f = (int)(idx & 63);
    float di = dinv[i];
    float v = AGG[idx] + di * di * XW[idx] + bias[f];
    H[idx] = softplus_f(v);
}

// y = h @ Wout   ([N,64] x [64,1])
__global__ void proj_kernel(const float* __restrict__ H,
                            const float* __restrict__ Wout,
                            float* __restrict__ Y, int N) {
    int i = blockIdx.x * blockDim.x + threadIdx.x;
    if (i >= N) return;
    const float* hp = H + (size_t)i * NF;
    float s = 0.0f;
#pragma unroll
    for (int k = 0; k < NF; ++k) s += hp[k] * Wout[k];
    Y[i] = s;
}

__global__ void scatter_out_kernel(const int* __restrict__ rows,
                                   const int* __restrict__ cols,
                                   const float* __restrict__ dinvO,
                                   const float* __restrict__ Y,
                                   float* __restrict__ OUT, int E) {
    int e = blockIdx.x * blockDim.x + threadIdx.x;
    if (e >= E) return;
    int r = rows[e];
    int c = cols[e];
    unsafeAtomicAdd(&OUT[c], dinvO[r] * dinvO[c] * Y[r]);
}

__global__ void finish_out_kernel(const float* __restrict__ dinvO,
                                  const float* __restrict__ Y,
                                  const float* __restrict__ bout,
                                  float* __restrict__ OUT, int N) {
    int i = blockIdx.x * blockDim.x + threadIdx.x;
    if (i >= N) return;
    float di = dinvO[i];
    OUT[i] = OUT[i] + di * di * Y[i] + bout[0];
}

// ---------------------------------------------------------------- driver
extern "C" void kernel_launch(void* const* d_in, const int* in_sizes, int n_in,
                              void* d_out, int out_size, void* d_ws, size_t ws_size,
                              hipStream_t stream) {
    const float* x    = (const float*)d_in[1];
    const float* dtr  = (const float*)d_in[2];
    const float* npd  = (const float*)d_in[3];
    const float* kp   = (const float*)d_in[4];
    const float* kpp  = (const float*)d_in[5];
    const float* imin = (const float*)d_in[6];
    const float* imax = (const float*)d_in[7];
    const int*   ei   = (const int*)d_in[8];
    const float* ed   = (const float*)d_in[9];
    const float* W1 = (const float*)d_in[10]; const float* b1 = (const float*)d_in[11];
    const float* W2 = (const float*)d_in[12]; const float* b2 = (const float*)d_in[13];
    const float* W3 = (const float*)d_in[14]; const float* b3 = (const float*)d_in[15];
    const float* Wo = (const float*)d_in[16]; const float* bo = (const float*)d_in[17];
    float* out = (float*)d_out;

    const int N = in_sizes[2];        // 100000
    const int E = in_sizes[9];        // 3200000
    const int* rows = ei;
    const int* cols = ei + E;

    // workspace carve-up (floats)
    float* ws    = (float*)d_ws;
    float* h0    = ws;  ws += (size_t)N * 8;
    float* dinvW = ws;  ws += N;
    float* dinvO = ws;  ws += N;
    float* xw    = ws;  ws += (size_t)N * NF;
    float* agg   = ws;  ws += (size_t)N * NF;
    float* h     = ws;  ws += (size_t)N * NF;
    float* y     = ws;  ws += N;

    const int B = 256;
    const int ntiles = N / 16;                       // 6250, exact
    dim3 gemmGrid((ntiles + 7) / 8);                 // 8 waves / block
    long long nf_total = (long long)N * NF;
    int gNF   = (int)((nf_total + B - 1) / B);
    int gN    = (N + B - 1) / B;
    int gE    = (E + B - 1) / B;
    long long sc_total = (long long)E * 16;
    int gSC   = (int)((sc_total + B - 1) / B);

    // degrees / normalization (shared by layers 1-3; dinvO for output conv)
    zero_kernel<<<(2 * N + B - 1) / B, B, 0, stream>>>(dinvW, 2 * N); // dinvW+dinvO adjacent
    build_h0_kernel<<<gN, B, 0, stream>>>(x, dtr, npd, kp, kpp, imin, imax, h0, N);
    deg_kernel<<<gE, B, 0, stream>>>(cols, ed, dinvW, dinvO, E);
    dinv_kernel<<<gN, B, 0, stream>>>(dinvW, dinvO, N);

    // ---- layer 1: [N,8] x [8,64]
    zero_kernel<<<gNF, B, 0, stream>>>(agg, nf_total);
    gemm_wmma_kernel<8><<<gemmGrid, B, 0, stream>>>(h0, W1, xw, ntiles);
    scatter_kernel<<<gSC, B, 0, stream>>>(rows, cols, ed, dinvW, xw, agg, E, 1);
    act_kernel<<<gNF, B, 0, stream>>>(agg, xw, dinvW, b1, h, N);

    // ---- layer 2
    zero_kernel<<<gNF, B, 0, stream>>>(agg, nf_total);
    gemm_wmma_kernel<64><<<gemmGrid, B, 0, stream>>>(h, W2, xw, ntiles);
    scatter_kernel<<<gSC, B, 0, stream>>>(rows, cols, ed, dinvW, xw, agg, E, 1);
    act_kernel<<<gNF, B, 0, stream>>>(agg, xw, dinvW, b2, h, N);

    // ---- layer 3
    zero_kernel<<<gNF, B, 0, stream>>>(agg, nf_total);
    gemm_wmma_kernel<64><<<gemmGrid, B, 0, stream>>>(h, W3, xw, ntiles);
    scatter_kernel<<<gSC, B, 0, stream>>>(rows, cols, ed, dinvW, xw, agg, E, 1);
    act_kernel<<<gNF, B, 0, stream>>>(agg, xw, dinvW, b3, h, N);

    // ---- output conv: unit edge weights, Wout [64,1]
    proj_kernel<<<gN, B, 0, stream>>>(h, Wo, y, N);
    zero_kernel<<<gN, B, 0, stream>>>(out, N);
    scatter_out_kernel<<<gE, B, 0, stream>>>(rows, cols, dinvO, y, out, E);
    finish_out_kernel<<<gN, B, 0, stream>>>(dinvO, y, bo, out, N);
}